// OVSSM_32160715112682
// MI455X (gfx1250) — compile-verified
//
#include <hip/hip_runtime.h>
#include <math.h>

// ---------------- problem constants ----------------
#define BSZ   2
#define HH    64
#define WW2   64
#define HW    4096          // 64*64
#define CM    96            // D_MODEL
#define DD    255           // D_INNER
#define DD2   510
#define NS    16            // D_STATE
#define RR    6             // DT_RANK
#define KK    4             // scan directions
#define LS    8192          // 2*H*W
#define XC    38            // DT_RANK + 2*D_STATE

typedef __attribute__((ext_vector_type(2))) float v2f;
typedef __attribute__((ext_vector_type(8))) float v8f;

#define AS1 __attribute__((address_space(1)))
#define AS3 __attribute__((address_space(3)))

#if __has_builtin(__builtin_amdgcn_global_load_async_to_lds_b32)
#define HAVE_ASYNC_LDS 1
#else
#define HAVE_ASYNC_LDS 0
#endif

__device__ __forceinline__ void wait_async_le(int n) {
#if __has_builtin(__builtin_amdgcn_s_wait_asynccnt)
  if (n == 0) __builtin_amdgcn_s_wait_asynccnt(0);
  else        __builtin_amdgcn_s_wait_asynccnt(1);
#else
  if (n == 0) asm volatile("s_wait_asynccnt 0x0" ::: "memory");
  else        asm volatile("s_wait_asynccnt 0x1" ::: "memory");
#endif
}

// scan index for m=0 slot; m=1 slot is l+64
__device__ __forceinline__ int scan_index(int k, int y, int x) {
  int hg = y >> 3, i = y & 7, wg = x >> 3, j = x & 7;
  switch (k) {
    case 0:  return (((hg << 3) + wg) << 7) + (i << 3) + j;                         // ltr_utd
    case 1:  return ((((7 - hg) << 3) + (7 - wg)) << 7) + ((7 - i) << 3) + (7 - j); // rtl_dtu
    case 2:  return (((wg << 3) + hg) << 7) + (j << 3) + i;                         // utd_ltr
    default: return ((((7 - wg) << 3) + (7 - hg)) << 7) + ((7 - j) << 3) + (7 - i); // dtu_rtl
  }
}

__device__ __forceinline__ float silu_f(float v) { return v / (1.0f + expf(-v)); }

// ---------------------------------------------------------------
// K1: input projection.  X:(8192,96) @ W^T(96->510)
//     cols [0,255)  -> xraw (B, DD, H, W);  cols [255,510) -> z (B, HW, DD)
// One wave -> one 16x16 tile, V_WMMA_F32_16X16X4_F32, K=96 (no tail).
// N-tail handled by address clamping: a WMMA B-column only feeds the same
// D-column, and out-of-range columns are never stored.
// ---------------------------------------------------------------
__global__ void k_inproj(const float* __restrict__ X, const float* __restrict__ Wt,
                         float* __restrict__ xraw, float* __restrict__ z) {
  int wave = threadIdx.x >> 5, lane = threadIdx.x & 31;
  int tile = blockIdx.x * 4 + wave;            // 512 M-tiles * 32 N-tiles
  int mt = tile >> 5, nt = tile & 31;
  int m0 = mt << 4, n0 = nt << 4;
  int half = lane >> 4, q = lane & 15;
  int arow = m0 + q;
  int bcol = n0 + q;
  int bsafe = (bcol < DD2) ? bcol : 0;         // clamp instead of branch
  const float* Xr = X + (size_t)arow * CM;
  const float* Wr = Wt + (size_t)bsafe * CM;
  v8f acc = {};
#pragma unroll 4
  for (int k = 0; k < CM; k += 4) {
    int kb = k + half * 2;
    v2f a, b;
    a.x = Xr[kb]; a.y = Xr[kb + 1];
    b.x = Wr[kb]; b.y = Wr[kb + 1];
    acc = __builtin_amdgcn_wmma_f32_16x16x4_f32(false, a, false, b, (short)0, acc,
                                                false, false);
  }
  int c = n0 + q;
  if (c >= DD2) return;
#pragma unroll
  for (int r = 0; r < 8; ++r) {
    int m = m0 + r + 8 * half;                 // global row in (B*HW)
    int b = m >> 12, p = m & 4095;
    float v = acc[r];
    if (c < DD) xraw[((size_t)(b * DD + c)) * HW + p] = v;
    else        z[((size_t)(b * HW + p)) * DD + (c - DD)] = v;
  }
}

// ---------------------------------------------------------------
// K2: depthwise 3x3 conv (pad 1) + bias + SiLU.  (B,DD,H,W) -> (B,DD,H,W)
// ---------------------------------------------------------------
__global__ void k_dwconv(const float* __restrict__ xr, const float* __restrict__ cw,
                         const float* __restrict__ cb, float* __restrict__ xo) {
  int idx = blockIdx.x * 256 + threadIdx.x;
  if (idx >= BSZ * DD * HW) return;
  int p = idx & 4095, cdb = idx >> 12;
  int c = cdb % DD;
  int y = p >> 6, x = p & 63;
  const float* src = xr + ((size_t)cdb) * HW;
  float s = cb[c];
#pragma unroll
  for (int dy = 0; dy < 3; ++dy) {
    int yy = y + dy - 1;
    if (yy < 0 || yy >= HH) continue;
#pragma unroll
    for (int dx = 0; dx < 3; ++dx) {
      int xx = x + dx - 1;
      if (xx < 0 || xx >= WW2) continue;
      s += src[yy * WW2 + xx] * cw[c * 9 + dy * 3 + dx];
    }
  }
  xo[idx] = silu_f(s);
}

// ---------------------------------------------------------------
// K3: build xs (B,K,L,DD)  (d fastest -> coalesced scan access)
// ---------------------------------------------------------------
__global__ void k_scatter(const float* __restrict__ xv, const float* __restrict__ xi,
                          float* __restrict__ xs) {
  int idx = blockIdx.x * 256 + threadIdx.x;
  if (idx >= BSZ * DD * HW) return;
  int p = idx & 4095, cdb = idx >> 12;
  int d = cdb % DD, b = cdb / DD;
  float v0 = xv[idx], v1 = xi[idx];
  int y = p >> 6, x = p & 63;
#pragma unroll
  for (int k = 0; k < KK; ++k) {
    int l0 = scan_index(k, y, x);
    size_t base = ((size_t)(b * KK + k)) * LS;
    xs[(base + l0) * DD + d]      = v0;
    xs[(base + l0 + 64) * DD + d] = v1;
  }
}

// ---------------------------------------------------------------
// K4: x_dbl[b,k,l,c] = sum_d x_proj_w[k,c,d] * xs[b,k,l,d]
// per (b,k): C(38 x 8192) = A(38 x 255) * B(255 x 8192), WMMA f32.
// K-loop: 63 unguarded steps + 1 guarded tail (K=255).
// A-row overhang clamped (rows >=38 are computed but never stored).
// ---------------------------------------------------------------
__global__ void k_xdbl(const float* __restrict__ xpw, const float* __restrict__ xs,
                       float* __restrict__ xdbl) {
  int wave = threadIdx.x >> 5, lane = threadIdx.x & 31;
  int tile = blockIdx.x * 4 + wave;            // (bk * 3 + mt) * 512 + nt
  int nt = tile & 511; tile >>= 9;
  int mt = tile % 3, bk = tile / 3;
  int k = bk & 3;
  int m0 = mt << 4, n0 = nt << 4;
  int half = lane >> 4, q = lane & 15;
  int arow = m0 + q;
  int arow_s = (arow < XC) ? arow : 0;         // clamp; rows >=38 never stored
  int nl = n0 + q;
  const float* Ar = xpw + (size_t)k * XC * DD + (size_t)arow_s * DD;  // (255)
  const float* Br = xs + ((size_t)bk * LS + nl) * DD;                 // (255)
  v8f acc = {};
#pragma unroll 4
  for (int kk = 0; kk < 252; kk += 4) {        // K 0..251, no guards
    int kb = kk + half * 2;
    v2f a, b;
    a.x = Ar[kb]; a.y = Ar[kb + 1];
    b.x = Br[kb]; b.y = Br[kb + 1];
    acc = __builtin_amdgcn_wmma_f32_16x16x4_f32(false, a, false, b, (short)0, acc,
                                                false, false);
  }
  {                                            // tail: K 252..254, pad 255 -> 0
    int kb = 252 + half * 2;
    v2f a, b;
    a.x = Ar[kb]; a.y = (half == 0) ? Ar[kb + 1] : 0.f;
    b.x = Br[kb]; b.y = (half == 0) ? Br[kb + 1] : 0.f;
    acc = __builtin_amdgcn_wmma_f32_16x16x4_f32(false, a, false, b, (short)0, acc,
                                                false, false);
  }
#pragma unroll
  for (int r = 0; r < 8; ++r) {
    int c = m0 + r + 8 * half;
    if (c < XC)
      xdbl[(((size_t)bk * LS) + nl) * XC + c] = acc[r];
  }
}

// ---------------------------------------------------------------
// K5: selective scan.  One block per (b,k); thread d owns channel d
// (h[16], A[16], dt_w[6] in VGPRs).  The 38-float (dt_low|B|C) column is
// double-buffered in LDS via async load-to-LDS: issue column l+1, then
// s_wait_asynccnt<=1 releases column l (async loads complete in order),
// overlapping the fetch with the 16-exp recurrence chain.
// y overwrites xs in place.
// ---------------------------------------------------------------
__global__ void k_scan(const float* __restrict__ xdbl, const float* __restrict__ dtw,
                       const float* __restrict__ dtb, const float* __restrict__ alog,
                       const float* __restrict__ Dv, float* __restrict__ xs) {
  int bk = blockIdx.x;
  int k = bk & 3;
  int d = threadIdx.x;
  bool act = d < DD;
  __shared__ float col[2][40];

  float wdt[RR], Aa[NS], h[NS];
  float bias = 0.f, Dc = 0.f;
  if (act) {
#pragma unroll
    for (int r = 0; r < RR; ++r) wdt[r] = dtw[(k * DD + d) * RR + r];
    bias = dtb[k * DD + d];
#pragma unroll
    for (int n = 0; n < NS; ++n) Aa[n] = -expf(alog[(k * DD + d) * NS + n]);
    Dc = Dv[k * DD + d];
  }
#pragma unroll
  for (int n = 0; n < NS; ++n) h[n] = 0.f;

  const float* xd = xdbl + (size_t)bk * LS * XC;     // (L,38)
  float* u = xs + (size_t)bk * LS * DD;              // (L,DD), in/out

#if HAVE_ASYNC_LDS
  if (threadIdx.x < XC)                              // prologue: column 0
    __builtin_amdgcn_global_load_async_to_lds_b32(
        (AS1 int*)(xd + threadIdx.x),
        (AS3 int*)&col[0][threadIdx.x], 0, 0);
#endif

  for (int l = 0; l < LS; ++l) {
#if HAVE_ASYNC_LDS
    if (l + 1 < LS) {
      if (threadIdx.x < XC)
        __builtin_amdgcn_global_load_async_to_lds_b32(
            (AS1 int*)(xd + (size_t)(l + 1) * XC + threadIdx.x),
            (AS3 int*)&col[(l + 1) & 1][threadIdx.x], 0, 0);
      wait_async_le(1);                              // column l complete
    } else {
      wait_async_le(0);
    }
#else
    if (threadIdx.x < XC)
      col[l & 1][threadIdx.x] = xd[(size_t)l * XC + threadIdx.x];
#endif
    __syncthreads();
    const float* cc = col[l & 1];
    if (act) {
      float dv = bias;
#pragma unroll
      for (int r = 0; r < RR; ++r) dv += cc[r] * wdt[r];
      float delta = (dv > 20.f) ? dv : log1pf(expf(dv));   // softplus
      size_t ui = (size_t)l * DD + d;
      float uu = u[ui];
      if (l + 8 < LS) __builtin_prefetch(&u[(size_t)(l + 8) * DD + d], 0, 1);
      float du = delta * uu;
      float acc = 0.f;
#pragma unroll
      for (int n = 0; n < NS; ++n) {
        h[n] = expf(delta * Aa[n]) * h[n] + du * cc[RR + n];
        acc += h[n] * cc[RR + NS + n];
      }
      u[ui] = acc + Dc * uu;                               // y in place
    }
    __syncthreads();
  }
}

// ---------------------------------------------------------------
// K6: inverse permutation + sum over 4 directions. yv/yi: (B,HW,DD)
// ---------------------------------------------------------------
__global__ void k_recon(const float* __restrict__ y, float* __restrict__ yv,
                        float* __restrict__ yi) {
  int idx = blockIdx.x * 256 + threadIdx.x;
  if (idx >= BSZ * HW * DD) return;
  int d = idx % DD, bp = idx / DD;
  int p = bp & 4095, b = bp >> 12;
  int py = p >> 6, px = p & 63;
  float sv = 0.f, si = 0.f;
#pragma unroll
  for (int k = 0; k < KK; ++k) {
    int l0 = scan_index(k, py, px);
    size_t base = ((size_t)(b * KK + k)) * LS;
    sv += y[(base + l0) * DD + d];
    si += y[(base + l0 + 64) * DD + d];
  }
  yv[idx] = sv;
  yi[idx] = si;
}

// ---------------------------------------------------------------
// K7a: LayerNorm over DD + SiLU gate, in place.  One wave per row.
// ---------------------------------------------------------------
__global__ void k_lngate(float* __restrict__ yv, const float* __restrict__ z,
                         const float* __restrict__ gw, const float* __restrict__ gb) {
  int wave = threadIdx.x >> 5, lane = threadIdx.x & 31;
  int row = blockIdx.x * 8 + wave;
  float* yr = yv + (size_t)row * DD;
  const float* zr = z + (size_t)row * DD;
  float vals[8];
  float s = 0.f, s2 = 0.f;
#pragma unroll
  for (int i = 0; i < 8; ++i) {
    int d = lane + 32 * i;
    float v = (d < DD) ? yr[d] : 0.f;
    vals[i] = v;
    s += v; s2 += v * v;
  }
#pragma unroll
  for (int off = 16; off >= 1; off >>= 1) {
    s  += __shfl_xor(s, off, 32);
    s2 += __shfl_xor(s2, off, 32);
  }
  float mean = s / (float)DD;
  float var  = s2 / (float)DD - mean * mean;
  float rstd = rsqrtf(var + 1e-5f);
#pragma unroll
  for (int i = 0; i < 8; ++i) {
    int d = lane + 32 * i;
    if (d < DD) {
      float g = silu_f(zr[d]);
      yr[d] = ((vals[i] - mean) * rstd * gw[d] + gb[d]) * g;
    }
  }
}

// ---------------------------------------------------------------
// K7b: output projection (8192 x 255) @ (255 -> 96), WMMA f32.
// 63 unguarded K-steps + guarded tail.
// ---------------------------------------------------------------
__global__ void k_outproj(const float* __restrict__ G, const float* __restrict__ Wo,
                          float* __restrict__ out) {
  int wave = threadIdx.x >> 5, lane = threadIdx.x & 31;
  int tile = blockIdx.x * 4 + wave;            // 512 M-tiles * 6 N-tiles
  int mt = tile / 6, nt = tile % 6;
  int m0 = mt << 4, n0 = nt << 4;
  int half = lane >> 4, q = lane & 15;
  int arow = m0 + q, bcol = n0 + q;            // always in range
  const float* Gr = G + (size_t)arow * DD;
  const float* Wr = Wo + (size_t)bcol * DD;
  v8f acc = {};
#pragma unroll 4
  for (int kk = 0; kk < 252; kk += 4) {
    int kb = kk + half * 2;
    v2f a, b;
    a.x = Gr[kb]; a.y = Gr[kb + 1];
    b.x = Wr[kb]; b.y = Wr[kb + 1];
    acc = __builtin_amdgcn_wmma_f32_16x16x4_f32(false, a, false, b, (short)0, acc,
                                                false, false);
  }
  {
    int kb = 252 + half * 2;
    v2f a, b;
    a.x = Gr[kb]; a.y = (half == 0) ? Gr[kb + 1] : 0.f;
    b.x = Wr[kb]; b.y = (half == 0) ? Wr[kb + 1] : 0.f;
    acc = __builtin_amdgcn_wmma_f32_16x16x4_f32(false, a, false, b, (short)0, acc,
                                                false, false);
  }
#pragma unroll
  for (int r = 0; r < 8; ++r) {
    int m = m0 + r + 8 * half;
    out[(size_t)m * CM + bcol] = acc[r];
  }
}

// ---------------------------------------------------------------
extern "C" void kernel_launch(void* const* d_in, const int* in_sizes, int n_in,
                              void* d_out, int out_size, void* d_ws, size_t ws_size,
                              hipStream_t stream) {
  const float* x_vis     = (const float*)d_in[0];
  const float* x_inf     = (const float*)d_in[1];
  const float* w_in_vis  = (const float*)d_in[2];
  const float* w_in_inf  = (const float*)d_in[3];
  const float* conv_w_v  = (const float*)d_in[4];
  const float* conv_b_v  = (const float*)d_in[5];
  const float* conv_w_i  = (const float*)d_in[6];
  const float* conv_b_i  = (const float*)d_in[7];
  const float* x_proj_w  = (const float*)d_in[8];
  const float* dt_w      = (const float*)d_in[9];
  const float* dt_b      = (const float*)d_in[10];
  const float* A_logs    = (const float*)d_in[11];
  const float* Ds        = (const float*)d_in[12];
  const float* ln_g_vis  = (const float*)d_in[13];
  const float* ln_b_vis  = (const float*)d_in[14];
  const float* ln_g_inf  = (const float*)d_in[15];
  const float* ln_b_inf  = (const float*)d_in[16];
  const float* w_out_vis = (const float*)d_in[17];
  const float* w_out_inf = (const float*)d_in[18];
  float* out = (float*)d_out;

  const size_t SZ1 = (size_t)BSZ * DD * HW;          // 2,088,960 floats
  float* ws    = (float*)d_ws;
  float* XVRAW = ws;                                  // (B,DD,H,W) pre-conv / later YV (B,HW,DD)
  float* XIRAW = ws + SZ1;
  float* ZV    = ws + 2 * SZ1;                        // (B,HW,DD)
  float* ZI    = ws + 3 * SZ1;
  float* XV    = ws + 4 * SZ1;                        // (B,DD,H,W) post conv+silu
  float* XI    = ws + 5 * SZ1;
  float* XS    = ws + 6 * SZ1;                        // (B,K,L,DD), scan output in place
  float* XDBL  = XS + (size_t)BSZ * KK * LS * DD;     // (B,K,L,38)
  float* YV = XVRAW;                                  // reuse
  float* YI = XIRAW;

  // 1) input projections  (512*32 tiles, 4 waves/block)
  k_inproj<<<4096, 128, 0, stream>>>(x_vis, w_in_vis, XVRAW, ZV);
  k_inproj<<<4096, 128, 0, stream>>>(x_inf, w_in_inf, XIRAW, ZI);

  // 2) depthwise conv + SiLU
  int nElem = BSZ * DD * HW;
  int nb = (nElem + 255) / 256;
  k_dwconv<<<nb, 256, 0, stream>>>(XVRAW, conv_w_v, conv_b_v, XV);
  k_dwconv<<<nb, 256, 0, stream>>>(XIRAW, conv_w_i, conv_b_i, XI);

  // 3) build 4-direction window scan sequences
  k_scatter<<<nb, 256, 0, stream>>>(XV, XI, XS);

  // 4) x_dbl projection  (8 bk * 3 Mtiles * 512 Ntiles, 4 waves/block)
  k_xdbl<<<3072, 128, 0, stream>>>(x_proj_w, XS, XDBL);

  // 5) selective scan (8 sequences, 255 channels each)
  k_scan<<<BSZ * KK, 256, 0, stream>>>(XDBL, dt_w, dt_b, A_logs, Ds, XS);

  // 6) reconstruct
  int nR = BSZ * HW * DD;
  k_recon<<<(nR + 255) / 256, 256, 0, stream>>>(XS, YV, YI);

  // 7) LN + gate, then output projections
  k_lngate<<<BSZ * HW / 8, 256, 0, stream>>>(YV, ZV, ln_g_vis, ln_b_vis);
  k_lngate<<<BSZ * HW / 8, 256, 0, stream>>>(YI, ZI, ln_g_inf, ln_b_inf);

  k_outproj<<<768, 128, 0, stream>>>(YV, w_out_vis, out);
  k_outproj<<<768, 128, 0, stream>>>(YI, w_out_inf, out + (size_t)BSZ * HW * CM);
}